// GRAPHPREPROCESS2_49503793054249
// MI455X (gfx1250) — compile-verified
//
#include <hip/hip_runtime.h>
#include <math.h>

#define BS    8192
#define IND   936
#define DV    64
#define NNODE 12
#define NK    15
#define KLOC  5
#define KTOP  3

typedef __attribute__((ext_vector_type(2))) float v2f;
typedef __attribute__((ext_vector_type(8))) float v8f;

// ---------------------------------------------------------------------------
// V_WMMA_F32_16X16X4_F32 wrapper.  D(16x16,f32) = A(16x4,f32) x B(4x16,f32) + C
// A frag: lane<16 holds A[lane][k0],A[lane][k0+1]; lane>=16 holds k0+2,k0+3
// B frag: lane<16 holds B[k0][lane&15],B[k0+1][..]; lane>=16 holds k0+2,k0+3
// C/D:    vgpr q: lanes0-15 -> (M=q, N=lane), lanes16-31 -> (M=q+8, N=lane-16)
// ---------------------------------------------------------------------------
__device__ __forceinline__ v8f wmma4(v2f a, v2f b, v8f c) {
  return __builtin_amdgcn_wmma_f32_16x16x4_f32(false, a, false, b, (short)0, c,
                                               false, false);
}

// ---------------------------------------------------------------------------
// WMMA linear: Y[M,N] = act(X[M,K] @ W[N,K]^T + bias).  Unguarded hot loop:
// requires M%16==0, K%4==0, (N/16)%NT==0.  One wave computes a 16 x (NT*16)
// strip so each A fragment (b64 load) feeds NT wmmas.
// act: 0=none, 1=LeakyReLU(0.01), 2=sigmoid
// ---------------------------------------------------------------------------
template <int NT>
__global__ void k_linear_t(const float* __restrict__ X, int lda,
                           const float* __restrict__ W,
                           const float* __restrict__ bias,
                           float* __restrict__ Y, int ldc,
                           int M, int N, int K, int act) {
  int gwave = (int)((blockIdx.x * (size_t)blockDim.x + threadIdx.x) >> 5);
  int lane  = threadIdx.x & 31;
  int ngrp  = (N >> 4) / NT;
  int mtiles = M >> 4;
  if (gwave >= mtiles * ngrp) return;
  int mt = gwave / ngrp, ng = gwave % ngrp;
  int m0 = mt << 4, n0 = ng * (NT * 16);
  int r = lane & 15, h = lane >> 4;
  const float* Xr = X + (size_t)(m0 + r) * lda + (h << 1);
  const float* Wr[NT];
#pragma unroll
  for (int t = 0; t < NT; ++t)
    Wr[t] = W + (size_t)(n0 + t * 16 + r) * K + (h << 1);
  v8f z = {};
  v8f acc[NT];
#pragma unroll
  for (int t = 0; t < NT; ++t) acc[t] = z;
  for (int k0 = 0; k0 < K; k0 += 4) {
    v2f a = *(const v2f*)(Xr + k0);
#pragma unroll
    for (int t = 0; t < NT; ++t) {
      v2f b = *(const v2f*)(Wr[t] + k0);
      acc[t] = wmma4(a, b, acc[t]);
    }
  }
#pragma unroll
  for (int t = 0; t < NT; ++t) {
    int bn = n0 + t * 16 + r;
    float bv = bias ? bias[bn] : 0.f;
#pragma unroll
    for (int q = 0; q < 8; ++q) {
      float v = acc[t][q] + bv;
      if (act == 1)      v = (v >= 0.f) ? v : 0.01f * v;
      else if (act == 2) v = 1.f / (1.f + __expf(-v));
      Y[(size_t)(m0 + (h << 3) + q) * ldc + bn] = v;
    }
  }
}

// ------------------- small helper kernels ----------------------------------
__global__ void k_zerof(float* p, int n) {
  for (int i = blockIdx.x * blockDim.x + threadIdx.x; i < n;
       i += gridDim.x * blockDim.x) p[i] = 0.f;
}

// pad W_adj (225x304) -> (240x304), b_adj -> 240
__global__ void k_padw(const float* __restrict__ W, const float* __restrict__ b,
                       float* __restrict__ Wp, float* __restrict__ bp) {
  int i = blockIdx.x * blockDim.x + threadIdx.x;
  if (i < 240 * 304) {
    int n = i / 304, k = i % 304;
    Wp[i] = (n < 225) ? W[(size_t)n * 304 + k] : 0.f;
  }
  if (i < 240) bp[i] = (i < 225) ? b[i] : 0.f;
}

// softmax over bs axis of scores.T -> w[NNODE][BS]  (with -inf -> -1e10)
__global__ void k_node_softmax(const float* __restrict__ sc, float* __restrict__ w) {
  int n = blockIdx.x, t = threadIdx.x;
  __shared__ float red[256];
  float m = -3.4e38f;
  for (int b = t; b < BS; b += 256) {
    float s = sc[(size_t)b * NNODE + n];
    if (isinf(s) && s < 0.f) s = -1e10f;
    m = fmaxf(m, s);
  }
  red[t] = m; __syncthreads();
  for (int o = 128; o > 0; o >>= 1) { if (t < o) red[t] = fmaxf(red[t], red[t + o]); __syncthreads(); }
  m = red[0]; __syncthreads();
  float l = 0.f;
  for (int b = t; b < BS; b += 256) {
    float s = sc[(size_t)b * NNODE + n];
    if (isinf(s) && s < 0.f) s = -1e10f;
    l += __expf(s - m);
  }
  red[t] = l; __syncthreads();
  for (int o = 128; o > 0; o >>= 1) { if (t < o) red[t] += red[t + o]; __syncthreads(); }
  l = red[0];
  float inv = 1.f / l;
  for (int b = t; b < BS; b += 256) {
    float s = sc[(size_t)b * NNODE + n];
    if (isinf(s) && s < 0.f) s = -1e10f;
    w[(size_t)n * BS + b] = __expf(s - m) * inv;
  }
}

// nf[n,d] = sum_b w[n,b]*emb[b,d]   (grid = NNODE*DV blocks)
__global__ void k_wsum(const float* __restrict__ w, const float* __restrict__ emb,
                       float* __restrict__ nf) {
  int n = blockIdx.x / DV, d = blockIdx.x % DV, t = threadIdx.x;
  __shared__ float red[256];
  float acc = 0.f;
  for (int b = t; b < BS; b += 256)
    acc += w[(size_t)n * BS + b] * emb[(size_t)b * DV + d];
  red[t] = acc; __syncthreads();
  for (int o = 128; o > 0; o >>= 1) { if (t < o) red[t] += red[t + o]; __syncthreads(); }
  if (t == 0) nf[n * DV + d] = red[0];
}

// row L2 norms of X[M,DV] (one wave32 per row)
__global__ void k_rownorm(const float* __restrict__ X, float* __restrict__ out, int M) {
  int wid = (int)((blockIdx.x * (size_t)blockDim.x + threadIdx.x) >> 5);
  int lane = threadIdx.x & 31;
  if (wid >= M) return;
  float v0 = X[(size_t)wid * DV + lane], v1 = X[(size_t)wid * DV + lane + 32];
  float s = v0 * v0 + v1 * v1;
  for (int o = 16; o > 0; o >>= 1) s += __shfl_xor(s, o, 32);
  if (lane == 0) out[wid] = sqrtf(s);
}

// alpha_p = softmax_b( cos(nf, emb) ) -> w1[NNODE][BS]
__global__ void k_cos_softmax_nodes(const float* __restrict__ nf, const float* __restrict__ nfn,
                                    const float* __restrict__ emb, const float* __restrict__ en,
                                    float* __restrict__ logits, float* __restrict__ w) {
  int n = blockIdx.x, t = threadIdx.x;
  __shared__ float nfv[DV];
  __shared__ float red[256];
  if (t < DV) nfv[t] = nf[n * DV + t];
  __syncthreads();
  float xn = nfn[n], m = -3.4e38f;
  for (int b = t; b < BS; b += 256) {
    const float* e = emb + (size_t)b * DV;
    float dot = 0.f;
#pragma unroll
    for (int d = 0; d < DV; ++d) dot += nfv[d] * e[d];
    float den = xn * en[b];
    float c = (den > 0.f) ? dot / den : 0.f;
    if (isnan(c)) c = 0.f;
    logits[(size_t)n * BS + b] = c;
    m = fmaxf(m, c);
  }
  red[t] = m; __syncthreads();
  for (int o = 128; o > 0; o >>= 1) { if (t < o) red[t] = fmaxf(red[t], red[t + o]); __syncthreads(); }
  m = red[0]; __syncthreads();
  float l = 0.f;
  for (int b = t; b < BS; b += 256) l += __expf(logits[(size_t)n * BS + b] - m);
  red[t] = l; __syncthreads();
  for (int o = 128; o > 0; o >>= 1) { if (t < o) red[t] += red[t + o]; __syncthreads(); }
  float inv = 1.f / red[0];
  for (int b = t; b < BS; b += 256)
    w[(size_t)n * BS + b] = __expf(logits[(size_t)n * BS + b] - m) * inv;
}

// beta_p[b,:] = softmax_n( cos(emb[b], nf2[n]) )
__global__ void k_betap(const float* __restrict__ emb, const float* __restrict__ en,
                        const float* __restrict__ nf2, const float* __restrict__ n2n,
                        float* __restrict__ beta) {
  int b = blockIdx.x * blockDim.x + threadIdx.x;
  if (b >= BS) return;
  float c[NNODE], m = -3.4e38f;
  const float* e = emb + (size_t)b * DV;
  for (int n = 0; n < NNODE; ++n) {
    float dot = 0.f;
    for (int d = 0; d < DV; ++d) dot += e[d] * nf2[n * DV + d];
    float den = en[b] * n2n[n];
    float v = (den > 0.f) ? dot / den : 0.f;
    if (isnan(v)) v = 0.f;
    c[n] = v; m = fmaxf(m, v);
  }
  float l = 0.f;
  for (int n = 0; n < NNODE; ++n) { c[n] = __expf(c[n] - m); l += c[n]; }
  float inv = 1.f / l;
  for (int n = 0; n < NNODE; ++n) beta[(size_t)b * NNODE + n] = c[n] * inv;
}

// t1[b,d] = sum_n beta[b,n]*nf2[n,d]
__global__ void k_betanf(const float* __restrict__ beta, const float* __restrict__ nf2,
                         float* __restrict__ t1) {
  int idx = blockIdx.x * blockDim.x + threadIdx.x;
  if (idx >= BS * DV) return;
  int d = idx % DV, b = idx / DV;
  float acc = 0.f;
  for (int n = 0; n < NNODE; ++n)
    acc += beta[(size_t)b * NNODE + n] * nf2[n * DV + d];
  t1[idx] = acc;
}

// g = emb - gfb, ghat = g / ||g|| (0 if norm 0).  One wave per row.
__global__ void k_g_and_norm(const float* __restrict__ emb, const float* __restrict__ gfb,
                             float* __restrict__ g, float* __restrict__ ghat) {
  int wid = (int)((blockIdx.x * (size_t)blockDim.x + threadIdx.x) >> 5);
  int lane = threadIdx.x & 31;
  if (wid >= BS) return;
  size_t base = (size_t)wid * DV;
  float v0 = emb[base + lane] - gfb[base + lane];
  float v1 = emb[base + lane + 32] - gfb[base + lane + 32];
  g[base + lane] = v0; g[base + lane + 32] = v1;
  float s = v0 * v0 + v1 * v1;
  for (int o = 16; o > 0; o >>= 1) s += __shfl_xor(s, o, 32);
  float nrm = sqrtf(s);
  float inv = (nrm > 0.f) ? 1.f / nrm : 0.f;
  ghat[base + lane] = v0 * inv; ghat[base + lane + 32] = v1 * inv;
}

// ---------------------------------------------------------------------------
// Streaming 8192x8192 Gram (cos(g,g)) with per-row top-5 (register-resident,
// branchless bubble insert + chunk-max prefilter) and diag capture.
// One wave per 16-row tile; 4 waves/block share the LDS-staged column chunk.
// ---------------------------------------------------------------------------
__global__ void k_gram_top5(const float* __restrict__ ghat, float* __restrict__ top5v,
                            int* __restrict__ top5i, float* __restrict__ diag) {
  int ws = threadIdx.x >> 5, lane = threadIdx.x & 31;
  int mt = blockIdx.x * 4 + ws, m0 = mt << 4;
  int r = lane & 15, h = lane >> 4;
  __shared__ float sB[16][68];          // staged ghat column chunk (16 x 64)
  __shared__ float sS[4][16][18];
  v2f afr[16];
#pragma unroll
  for (int kk = 0; kk < 16; ++kk)
    afr[kk] = *(const v2f*)(ghat + (size_t)(m0 + r) * DV + kk * 4 + (h << 1));
  float tV[KLOC]; int tI[KLOC];
#pragma unroll
  for (int s = 0; s < KLOC; ++s) { tV[s] = -3.4e38f; tI[s] = 0; }
  float diagv = 0.f;
  int srow = threadIdx.x >> 3, scol = (threadIdx.x & 7) << 3;
  for (int c = 0; c < BS / 16; ++c) {
    int n0 = c << 4;
    {
      const float* src = ghat + (size_t)(n0 + srow) * DV + scol;
      *(float4*)&sB[srow][scol]     = *(const float4*)(src);
      *(float4*)&sB[srow][scol + 4] = *(const float4*)(src + 4);
    }
    __syncthreads();
    v8f acc = {};
#pragma unroll
    for (int kk = 0; kk < 16; ++kk) {
      v2f b = *(const v2f*)&sB[r][kk * 4 + (h << 1)];
      acc = wmma4(afr[kk], b, acc);
    }
#pragma unroll
    for (int q = 0; q < 8; ++q) sS[ws][(h << 3) + q][r] = acc[q];
    __syncthreads();
    if (lane < 16) {
      float vals[16];
#pragma unroll
      for (int jj = 0; jj < 8; ++jj) {
        v2f p = *(const v2f*)&sS[ws][lane][jj * 2];
        vals[jj * 2] = p.x; vals[jj * 2 + 1] = p.y;
      }
      int dj = (m0 + lane) - n0;               // diagonal inside this chunk?
      if ((unsigned)dj < 16u) { diagv = vals[dj]; vals[dj] = 0.f; }
      float cmx = vals[0];
#pragma unroll
      for (int j = 1; j < 16; ++j) cmx = fmaxf(cmx, vals[j]);
      if (cmx > tV[KLOC - 1]) {                // chunk can contribute
#pragma unroll
        for (int j = 0; j < 16; ++j) {
          float nv = vals[j]; int ni = n0 + j;
          if (nv > tV[KLOC - 1]) {             // branchless bubble-through
#pragma unroll
            for (int s = 0; s < KLOC; ++s) {
              bool gt = nv > tV[s];
              float tv = tV[s]; int ti = tI[s];
              tV[s] = gt ? nv : tv; tI[s] = gt ? ni : ti;
              nv    = gt ? tv : nv; ni    = gt ? ti : ni;
            }
          }
        }
      }
    }
    __syncthreads();
  }
  if (lane < 16) {
    int gr = m0 + lane;
    diag[gr] = diagv;
#pragma unroll
    for (int s = 0; s < KLOC; ++s) {
      top5v[gr * KLOC + s] = tV[s];
      top5i[gr * KLOC + s] = tI[s];
    }
  }
}

// sparse ah^T @ g scatter: one wave per (row,slot)
__global__ void k_scatter(const float* __restrict__ top5v, const int* __restrict__ top5i,
                          const float* __restrict__ g, float* __restrict__ hidden,
                          float* __restrict__ colsum) {
  int wid = (int)((blockIdx.x * (size_t)blockDim.x + threadIdx.x) >> 5);
  int lane = threadIdx.x & 31;
  if (wid >= BS * KLOC) return;
  int i = wid / KLOC, s = wid % KLOC;
  float v = top5v[i * KLOC + s];
  int j = top5i[i * KLOC + s];
  if (lane == 0) atomicAdd(&colsum[j], v);
  atomicAdd(&hidden[(size_t)j * DV + lane],      v * g[(size_t)i * DV + lane]);
  atomicAdd(&hidden[(size_t)j * DV + lane + 32], v * g[(size_t)i * DV + lane + 32]);
}

// hidden[j] += (colsum[j]!=0)*diag[j]*g[j];  flag[j] = (rowsum != 0)
__global__ void k_diag_flag(const float* __restrict__ colsum, const float* __restrict__ diag,
                            const float* __restrict__ g, float* __restrict__ hidden,
                            int* __restrict__ flag) {
  int wid = (int)((blockIdx.x * (size_t)blockDim.x + threadIdx.x) >> 5);
  int lane = threadIdx.x & 31;
  if (wid >= BS) return;
  float add = (colsum[wid] != 0.f) ? diag[wid] : 0.f;
  size_t base = (size_t)wid * DV;
  float h0 = hidden[base + lane]      + add * g[base + lane];
  float h1 = hidden[base + lane + 32] + add * g[base + lane + 32];
  hidden[base + lane] = h0; hidden[base + lane + 32] = h1;
  float s = h0 + h1;
  for (int o = 16; o > 0; o >>= 1) s += __shfl_xor(s, o, 32);
  if (lane == 0) flag[wid] = (s != 0.f) ? 1 : 0;
}

// stable exclusive scan of flags (single block, 1024 threads, 8 elems each)
__global__ void k_scan(const int* __restrict__ flag, int* __restrict__ prefix,
                       int* __restrict__ nhid) {
  __shared__ int part[1024];
  int t = threadIdx.x, base = t * 8;
  int loc[8], s = 0;
#pragma unroll
  for (int i = 0; i < 8; ++i) { loc[i] = s; s += flag[base + i]; }
  part[t] = s; __syncthreads();
  for (int o = 1; o < 1024; o <<= 1) {
    int v = part[t];
    int add = (t >= o) ? part[t - o] : 0;
    __syncthreads();
    part[t] = v + add;
    __syncthreads();
  }
  int off = (t > 0) ? part[t - 1] : 0;
#pragma unroll
  for (int i = 0; i < 8; ++i) prefix[base + i] = off + loc[i];
  if (t == 1023) *nhid = part[1023];
}

__global__ void k_compact(const float* __restrict__ hidden, const int* __restrict__ flag,
                          const int* __restrict__ prefix, float* __restrict__ hidc) {
  int wid = (int)((blockIdx.x * (size_t)blockDim.x + threadIdx.x) >> 5);
  int lane = threadIdx.x & 31;
  if (wid >= BS || !flag[wid]) return;
  size_t dst = (size_t)prefix[wid] * DV, src = (size_t)wid * DV;
  hidc[dst + lane] = hidden[src + lane];
  hidc[dst + lane + 32] = hidden[src + lane + 32];
}

__global__ void k_hhat(const float* __restrict__ hidc, const int* __restrict__ nhid,
                       float* __restrict__ hhat) {
  int wid = (int)((blockIdx.x * (size_t)blockDim.x + threadIdx.x) >> 5);
  int lane = threadIdx.x & 31;
  if (wid >= *nhid) return;
  size_t base = (size_t)wid * DV;
  float v0 = hidc[base + lane], v1 = hidc[base + lane + 32];
  float s = v0 * v0 + v1 * v1;
  for (int o = 16; o > 0; o >>= 1) s += __shfl_xor(s, o, 32);
  float nrm = sqrtf(s);
  float inv = (nrm > 0.f) ? 1.f / nrm : 0.f;
  hhat[base + lane] = v0 * inv; hhat[base + lane + 32] = v1 * inv;
}

// zero hidc/hhat rows in [n_hidden, BS) so the flash kernel needs no guards
__global__ void k_padzero(const int* __restrict__ nhid, float* __restrict__ hidc,
                          float* __restrict__ hhat) {
  int wid = (int)((blockIdx.x * (size_t)blockDim.x + threadIdx.x) >> 5);
  int lane = threadIdx.x & 31;
  if (wid >= BS || wid < *nhid) return;
  size_t base = (size_t)wid * DV;
  hidc[base + lane] = 0.f; hidc[base + lane + 32] = 0.f;
  hhat[base + lane] = 0.f; hhat[base + lane + 32] = 0.f;
}

// ---------------------------------------------------------------------------
// Fused "flash" kernel: S = ghat @ hhat^T (cos), online softmax over n_hidden
// with register-resident m/l and top-3, out = softmax(S) @ hidden.
// Column chunks LDS-staged per block: hhat row-major (S B-frags), hidc
// transposed (P@H B-frags as b64).  Per-row rescale broadcast via shfl.
// ---------------------------------------------------------------------------
__global__ void k_flash(const float* __restrict__ ghat, const float* __restrict__ hhat,
                        const float* __restrict__ hidc, const int* __restrict__ nhidp,
                        float* __restrict__ out, float* __restrict__ top3w,
                        int* __restrict__ top3i) {
  int ws = threadIdx.x >> 5, lane = threadIdx.x & 31;
  int mt = blockIdx.x * 4 + ws, m0 = mt << 4;
  int r = lane & 15, h = lane >> 4;
  int nh = *nhidp;
  __shared__ float sHB[16][68];         // staged hhat chunk (rows = hidden idx)
  __shared__ float sVT[64][18];         // staged hidc chunk, transposed [d][j]
  __shared__ float sS[4][16][18];
  __shared__ float sP[4][16][18];
  v2f afr[16];
#pragma unroll
  for (int kk = 0; kk < 16; ++kk)
    afr[kk] = *(const v2f*)(ghat + (size_t)(m0 + r) * DV + kk * 4 + (h << 1));
  float tV[KTOP]; int tI[KTOP];
#pragma unroll
  for (int s = 0; s < KTOP; ++s) { tV[s] = -3.4e38f; tI[s] = 0; }
  float mrow = -3.4e38f, lrow = 0.f;
  v8f z = {};
  v8f acc[4] = {z, z, z, z};
  int nchunk = (nh + 15) >> 4;
  int srow = threadIdx.x >> 3, scol = (threadIdx.x & 7) << 3;
  for (int c = 0; c < nchunk; ++c) {
    int n0 = c << 4;
    {
      const float* sh = hhat + (size_t)(n0 + srow) * DV + scol;
      *(float4*)&sHB[srow][scol]     = *(const float4*)(sh);
      *(float4*)&sHB[srow][scol + 4] = *(const float4*)(sh + 4);
      const float* sv = hidc + (size_t)(n0 + srow) * DV + scol;
      float4 a0 = *(const float4*)(sv);
      float4 a1 = *(const float4*)(sv + 4);
      sVT[scol + 0][srow] = a0.x; sVT[scol + 1][srow] = a0.y;
      sVT[scol + 2][srow] = a0.z; sVT[scol + 3][srow] = a0.w;
      sVT[scol + 4][srow] = a1.x; sVT[scol + 5][srow] = a1.y;
      sVT[scol + 6][srow] = a1.z; sVT[scol + 7][srow] = a1.w;
    }
    __syncthreads();
    v8f s8 = {};
#pragma unroll
    for (int kk = 0; kk < 16; ++kk) {
      v2f b = *(const v2f*)&sHB[r][kk * 4 + (h << 1)];
      s8 = wmma4(afr[kk], b, s8);
    }
#pragma unroll
    for (int q = 0; q < 8; ++q) sS[ws][(h << 3) + q][r] = s8[q];
    __syncthreads();
    float scale = 0.f;
    if (lane < 16) {
      float vals[16];
#pragma unroll
      for (int jj = 0; jj < 8; ++jj) {
        v2f p = *(const v2f*)&sS[ws][lane][jj * 2];
        vals[jj * 2] = p.x; vals[jj * 2 + 1] = p.y;
      }
      int lim = nh - n0;                    // mask columns >= n_hidden
#pragma unroll
      for (int j = 0; j < 16; ++j) vals[j] = (j < lim) ? vals[j] : -3.4e38f;
      float cmx = vals[0];
#pragma unroll
      for (int j = 1; j < 16; ++j) cmx = fmaxf(cmx, vals[j]);
      if (cmx > tV[KTOP - 1]) {             // top-3 prefilter
#pragma unroll
        for (int j = 0; j < 16; ++j) {
          float nv = vals[j]; int ni = n0 + j;
          if (nv > tV[KTOP - 1]) {
#pragma unroll
            for (int s = 0; s < KTOP; ++s) {
              bool gt = nv > tV[s];
              float tv = tV[s]; int ti = tI[s];
              tV[s] = gt ? nv : tv; tI[s] = gt ? ni : ti;
              nv    = gt ? tv : nv; ni    = gt ? ti : ni;
            }
          }
        }
      }
      float cm = fmaxf(mrow, cmx);
      scale = __expf(mrow - cm);
      float l = lrow * scale;
      float prow[16];
#pragma unroll
      for (int j = 0; j < 16; ++j) { prow[j] = __expf(vals[j] - cm); l += prow[j]; }
#pragma unroll
      for (int jj = 0; jj < 8; ++jj) {
        v2f p; p.x = prow[jj * 2]; p.y = prow[jj * 2 + 1];
        *(v2f*)&sP[ws][lane][jj * 2] = p;
      }
      mrow = cm; lrow = l;
    }
    // broadcast per-row rescale factors to the whole wave (no LDS round trip)
    float fq[8];
#pragma unroll
    for (int q = 0; q < 8; ++q) fq[q] = __shfl(scale, (h << 3) + q, 32);
#pragma unroll
    for (int t = 0; t < 4; ++t)
#pragma unroll
      for (int q = 0; q < 8; ++q) acc[t][q] *= fq[q];
    __syncthreads();
#pragma unroll
    for (int t = 0; t < 4; ++t) {
#pragma unroll
      for (int ks = 0; ks < 4; ++ks) {
        int k = ks * 4 + (h << 1);
        v2f a = *(const v2f*)&sP[ws][r][k];
        v2f b = *(const v2f*)&sVT[t * 16 + r][k];
        acc[t] = wmma4(a, b, acc[t]);
      }
    }
    __syncthreads();
  }
  float invl = (lrow > 0.f) ? 1.f / lrow : 0.f;   // valid on lanes < 16
  float iq[8];
#pragma unroll
  for (int q = 0; q < 8; ++q) iq[q] = __shfl(invl, (h << 3) + q, 32);
#pragma unroll
  for (int t = 0; t < 4; ++t)
#pragma unroll
    for (int q = 0; q < 8; ++q)
      out[(size_t)(m0 + (h << 3) + q) * DV + t * 16 + r] = acc[t][q] * iq[q];
  if (lane < 16) {
    int gr = m0 + lane;
#pragma unroll
    for (int s = 0; s < KTOP; ++s) {
      top3w[gr * KTOP + s] = __expf(tV[s] - mrow) * invl;
      top3i[gr * KTOP + s] = tI[s];
    }
  }
}

// ------------------- output assembly kernels -------------------------------
__global__ void k_allnode(const float* __restrict__ beta, const float* __restrict__ nf2,
                          const float* __restrict__ t3w, const int* __restrict__ t3i,
                          const float* __restrict__ hidc, float* __restrict__ out0) {
  size_t idx = blockIdx.x * (size_t)blockDim.x + threadIdx.x;
  if (idx >= (size_t)BS * NK * DV) return;
  int d = idx % DV;
  size_t bs_ = idx / DV;
  int slot = bs_ % NK;
  int b = (int)(bs_ / NK);
  float v;
  if (slot < NNODE) v = beta[(size_t)b * NNODE + slot] * nf2[slot * DV + d];
  else {
    int k = slot - NNODE;
    v = t3w[b * KTOP + k] * hidc[(size_t)t3i[b * KTOP + k] * DV + d];
  }
  out0[idx] = v;
}

__global__ void k_len(float* __restrict__ out) {
  int i = blockIdx.x * blockDim.x + threadIdx.x;
  if (i < BS * NK) out[i] = (float)(i / NK);
}

__global__ void k_add(const float* a, const float* b, float* c, int n) {
  int i = blockIdx.x * blockDim.x + threadIdx.x;
  if (i < n) c[i] = a[i] + b[i];
}
__global__ void k_sub(const float* a, const float* b, float* c, int n) {
  int i = blockIdx.x * blockDim.x + threadIdx.x;
  if (i < n) c[i] = a[i] - b[i];
}

// copy mn (BS x 240) into adjin columns 64..303 (row stride 304)
__global__ void k_copymn(const float* __restrict__ mn, float* __restrict__ adjin) {
  int i = blockIdx.x * blockDim.x + threadIdx.x;
  if (i >= BS * 240) return;
  int b = i / 240, c = i % 240;
  adjin[(size_t)b * 304 + 64 + c] = mn[i];
}

// symmetrize + zero diagonal; adjraw has row stride 240 (padded N)
__global__ void k_adjsym(const float* __restrict__ a, float* __restrict__ out) {
  int idx = blockIdx.x * blockDim.x + threadIdx.x;
  if (idx >= BS * NK * NK) return;
  int rr = idx % (NK * NK);
  int b = idx / (NK * NK);
  int i = rr / NK, j = rr % NK;
  out[idx] = (i == j) ? 0.f
                      : 0.5f * (a[(size_t)b * 240 + i * NK + j] +
                                a[(size_t)b * 240 + j * NK + i]);
}

// ---------------------------------------------------------------------------
static inline void launch_linear(const float* X, int lda, const float* W,
                                 const float* bias, float* Y, int ldc, int M,
                                 int N, int K, int act, hipStream_t s) {
  int ntiles = N >> 4, mtiles = M >> 4;
  if ((ntiles & 3) == 0) {
    long long waves = (long long)mtiles * (ntiles >> 2);
    k_linear_t<4><<<(int)((waves + 7) / 8), 256, 0, s>>>(X, lda, W, bias, Y, ldc, M, N, K, act);
  } else if (ntiles % 5 == 0) {
    long long waves = (long long)mtiles * (ntiles / 5);
    k_linear_t<5><<<(int)((waves + 7) / 8), 256, 0, s>>>(X, lda, W, bias, Y, ldc, M, N, K, act);
  } else {
    long long waves = (long long)mtiles * ntiles;
    k_linear_t<1><<<(int)((waves + 7) / 8), 256, 0, s>>>(X, lda, W, bias, Y, ldc, M, N, K, act);
  }
}

extern "C" void kernel_launch(void* const* d_in, const int* in_sizes, int n_in,
                              void* d_out, int out_size, void* d_ws, size_t ws_size,
                              hipStream_t stream) {
  (void)in_sizes; (void)n_in; (void)out_size; (void)ws_size;
  const float* gf     = (const float*)d_in[0];
  const float* scores = (const float*)d_in[1];
  const float* W_emb  = (const float*)d_in[2];  const float* b_emb  = (const float*)d_in[3];
  const float* W_ps   = (const float*)d_in[4];  const float* b_ps   = (const float*)d_in[5];
  const float* W_psb  = (const float*)d_in[6];  const float* b_psb  = (const float*)d_in[7];
  const float* W_hs   = (const float*)d_in[8];  const float* b_hs   = (const float*)d_in[9];
  const float* W_hsb  = (const float*)d_in[10]; const float* b_hsb  = (const float*)d_in[11];
  const float* W_indi = (const float*)d_in[12]; const float* b_indi = (const float*)d_in[13];
  const float* W_mn   = (const float*)d_in[14]; const float* b_mn   = (const float*)d_in[15];
  const float* W_mg   = (const float*)d_in[16]; const float* b_mg   = (const float*)d_in[17];
  const float* W_adj  = (const float*)d_in[18]; const float* b_adj  = (const float*)d_in[19];
  float* outp = (float*)d_out;

  // workspace layout
  float* Wsp = (float*)d_ws;
  size_t off = 0;
  auto alloc = [&](size_t n) { float* p = Wsp + off; off += n; return p; };
  float* emb    = alloc((size_t)BS * DV);
  float* w0     = alloc((size_t)NNODE * BS);
  float* nf     = alloc(NNODE * DV);
  float* nfn    = alloc(16);
  float* en     = alloc(BS);
  float* logits = alloc((size_t)NNODE * BS);
  float* w1     = alloc((size_t)NNODE * BS);
  float* nf2    = alloc(NNODE * DV);
  float* nf2n   = alloc(16);
  float* betap  = alloc((size_t)BS * NNODE);
  float* t1     = alloc((size_t)BS * DV);
  float* t2     = alloc((size_t)BS * DV);
  float* gfb    = alloc((size_t)BS * DV);
  float* g      = alloc((size_t)BS * DV);
  float* ghat   = alloc((size_t)BS * DV);
  float* top5v  = alloc((size_t)BS * KLOC);
  float* diag   = alloc(BS);
  float* colsum = alloc(BS);
  float* hidden = alloc((size_t)BS * DV);
  float* hidc   = alloc((size_t)BS * DV);
  float* hhat   = alloc((size_t)BS * DV);
  float* g2pre  = alloc((size_t)BS * DV);
  float* top3w  = alloc((size_t)BS * KTOP);
  float* gfb2   = alloc((size_t)BS * DV);
  float* mnbuf  = alloc((size_t)BS * 240);
  float* adjin  = alloc((size_t)BS * 304);
  float* adjraw = alloc((size_t)BS * 240);
  float* W_adjp = alloc((size_t)240 * 304);
  float* b_adjp = alloc(240);
  int* top5i  = (int*)alloc((size_t)BS * KLOC);
  int* flag   = (int*)alloc(BS);
  int* prefix = (int*)alloc(BS);
  int* top3i  = (int*)alloc((size_t)BS * KTOP);
  int* nhid   = (int*)alloc(4);
  float* t3   = t1;   // reuse
  float* sgg  = t2;   // gfb2 + gfb
  float* dif  = t1;   // g - gfb2

  // ---- prefix: emb, nf, alpha, nf2, beta_p, gfb, g ----
  launch_linear(gf, IND, W_emb, b_emb, emb, DV, BS, DV, IND, 0, stream);
  k_padw<<<(240 * 304 + 255) / 256, 256, 0, stream>>>(W_adj, b_adj, W_adjp, b_adjp);
  k_node_softmax<<<NNODE, 256, 0, stream>>>(scores, w0);
  k_wsum<<<NNODE * DV, 256, 0, stream>>>(w0, emb, nf);
  k_rownorm<<<(BS * 32 + 255) / 256, 256, 0, stream>>>(emb, en, BS);
  k_rownorm<<<2, 256, 0, stream>>>(nf, nfn, NNODE);
  k_cos_softmax_nodes<<<NNODE, 256, 0, stream>>>(nf, nfn, emb, en, logits, w1);
  k_wsum<<<NNODE * DV, 256, 0, stream>>>(w1, emb, nf2);
  k_rownorm<<<2, 256, 0, stream>>>(nf2, nf2n, NNODE);
  k_betap<<<BS / 256, 256, 0, stream>>>(emb, en, nf2, nf2n, betap);
  k_betanf<<<(BS * DV) / 256, 256, 0, stream>>>(betap, nf2, t1);
  launch_linear(t1, DV, W_ps, b_ps, t2, DV, BS, DV, DV, 0, stream);
  launch_linear(t2, DV, W_psb, b_psb, gfb, DV, BS, DV, DV, 0, stream);
  k_g_and_norm<<<(BS * 32) / 256, 256, 0, stream>>>(emb, gfb, g, ghat);

  // ---- hidden_full: Gram + top5 + sparse scatter + diag + compaction ----
  k_gram_top5<<<BS / 64, 128, 0, stream>>>(ghat, top5v, top5i, diag);
  k_zerof<<<256, 256, 0, stream>>>(hidden, BS * DV);
  k_zerof<<<32, 256, 0, stream>>>(colsum, BS);
  k_scatter<<<(BS * KLOC * 32) / 256, 256, 0, stream>>>(top5v, top5i, g, hidden, colsum);
  k_diag_flag<<<(BS * 32) / 256, 256, 0, stream>>>(colsum, diag, g, hidden, flag);
  k_scan<<<1, 1024, 0, stream>>>(flag, prefix, nhid);
  k_compact<<<(BS * 32) / 256, 256, 0, stream>>>(hidden, flag, prefix, hidc);
  k_hhat<<<(BS * 32) / 256, 256, 0, stream>>>(hidc, nhid, hhat);
  k_padzero<<<(BS * 32) / 256, 256, 0, stream>>>(nhid, hidc, hhat);

  // ---- fused beta_h softmax / top-3 / beta_h @ hidden ----
  k_flash<<<BS / 64, 128, 0, stream>>>(ghat, hhat, hidc, nhid, g2pre, top3w, top3i);
  launch_linear(g2pre, DV, W_hs, b_hs, t3, DV, BS, DV, DV, 0, stream);
  launch_linear(t3, DV, W_hsb, b_hsb, gfb2, DV, BS, DV, DV, 0, stream);

  // ---- outputs ----
  size_t o0 = 0;
  size_t o1 = o0 + (size_t)BS * NK * DV;
  size_t o2 = o1 + (size_t)BS * NK;
  size_t o3 = o2 + (size_t)BS * NK * NK;
  k_allnode<<<(int)(((size_t)BS * NK * DV + 255) / 256), 256, 0, stream>>>(
      betap, nf2, top3w, top3i, hidc, outp + o0);
  k_len<<<(BS * NK + 255) / 256, 256, 0, stream>>>(outp + o1);
  k_add<<<(BS * DV + 255) / 256, 256, 0, stream>>>(gfb2, gfb, sgg, BS * DV);
  launch_linear(sgg, DV, W_mg, b_mg, adjin, 304, BS, DV, DV, 0, stream);           // cols 0..63
  launch_linear(outp + o0, DV, W_mn, b_mn, mnbuf, 16, BS * NK, 16, DV, 0, stream); // (bs*15,16)
  k_copymn<<<(BS * 240 + 255) / 256, 256, 0, stream>>>(mnbuf, adjin);              // cols 64..303
  launch_linear(adjin, 304, W_adjp, b_adjp, adjraw, 240, BS, 240, 304, 2, stream); // padded head
  k_adjsym<<<(BS * NK * NK + 255) / 256, 256, 0, stream>>>(adjraw, outp + o2);
  k_sub<<<(BS * DV + 255) / 256, 256, 0, stream>>>(g, gfb2, dif, BS * DV);
  launch_linear(dif, DV, W_indi, b_indi, outp + o3, DV, BS, DV, DV, 1, stream);    // leaky
}